// RecurCluster_86354612453684
// MI455X (gfx1250) — compile-verified
//
#include <hip/hip_runtime.h>

typedef __attribute__((ext_vector_type(2))) float v2f;
typedef __attribute__((ext_vector_type(8))) float v8f;

#define EMBED      64
#define K_CL       10
#define ITERS      10
#define SUMSTRIDE  68   // LDS row padding: bank = (4*idx + k) % 64 -> clusters spread over banks

// ws layout (floats):
//   [0, 1024)     padded centroids 16x64 (rows 10..15 = 0)
//   [1024, 1040)  cnorm[16] (3e38 for padded rows)
//   [1040, 1680)  global sums 10x64
//   [1680, 1690)  global counts[10]
#define WS_CENT  0
#define WS_CNORM 1024
#define WS_SUM   1040
#define WS_CNT   1680

__global__ void kmeans_init(const float* __restrict__ c0, float* __restrict__ ws) {
    int t = threadIdx.x;  // 1024 threads, 1 block
    float* cent  = ws + WS_CENT;
    float* cnorm = ws + WS_CNORM;
    float* gsum  = ws + WS_SUM;
    float* gcnt  = ws + WS_CNT;
    cent[t] = (t < K_CL * EMBED) ? c0[t] : 0.0f;
    if (t < K_CL * EMBED) gsum[t] = 0.0f;
    if (t < K_CL)         gcnt[t] = 0.0f;
    __syncthreads();
    if (t < 16) {
        float s = 0.0f;
        for (int k = 0; k < EMBED; ++k) { float v = cent[t * EMBED + k]; s = fmaf(v, v, s); }
        cnorm[t] = (t < K_CL) ? s : 3.0e38f;
    }
}

__global__ void __launch_bounds__(256) kmeans_assign(const float* __restrict__ pts,
                                                     float* __restrict__ ws, int ntiles) {
    __shared__ float lsum[K_CL * SUMSTRIDE];
    __shared__ float lcnt[K_CL];
    for (int i = threadIdx.x; i < K_CL * SUMSTRIDE; i += blockDim.x) lsum[i] = 0.0f;
    if (threadIdx.x < K_CL) lcnt[threadIdx.x] = 0.0f;
    __syncthreads();

    const float* cent  = ws + WS_CENT;
    const float* cnorm = ws + WS_CNORM;

    const int lane   = threadIdx.x & 31;
    const int wave   = threadIdx.x >> 5;
    const int gwave  = blockIdx.x * (blockDim.x >> 5) + wave;
    const int nwaves = gridDim.x * (blockDim.x >> 5);

    const int m  = lane & 15;   // A-row / point-within-tile for this lane
    const int hi = lane >> 4;   // half-wave selector

    // A fragments: centroids 16x64 as 16 slices of 16x4 f32.
    // A 16x4 layout: lanes 0-15 = rows, VGPR0/1 = K{0,1}; lanes 16-31 = K{2,3}.
    v2f a[16];
    #pragma unroll
    for (int j = 0; j < 16; ++j)
        a[j] = *(const v2f*)(cent + m * EMBED + 4 * j + 2 * hi);
    float cn[8];
    #pragma unroll
    for (int r = 0; r < 8; ++r) cn[r] = cnorm[r + 8 * hi];

    for (int t = gwave; t < ntiles; t += nwaves) {
        const float* pbase = pts + (size_t)(t * 16 + m) * EMBED + 2 * hi;
        if (t + nwaves < ntiles)  // wave-uniform
            __builtin_prefetch((const void*)(pbase + (size_t)nwaves * 16 * EMBED), 0, 3);

        // B fragments: pts^T 64x16; B 4x16 layout mirrors A: lane = col (point),
        // lanes 0-15 hold K{0,1}, lanes 16-31 hold K{2,3} -> contiguous float2 loads.
        v2f b[16];
        #pragma unroll
        for (int j = 0; j < 16; ++j)
            b[j] = *(const v2f*)(pbase + 4 * j);

        // G[16 centroids x 16 points] on the matrix pipe, fp32 K=4, 16 chained WMMAs
        v8f acc = {};
        #pragma unroll
        for (int j = 0; j < 16; ++j)
            acc = __builtin_amdgcn_wmma_f32_16x16x4_f32(
                false, a[j], false, b[j], (short)0, acc, false, false);

        // score[m'] = cnorm[m'] - 2*dot ; argmin in-lane over 8 centroids
        float best = 3.0e38f; int bidx = 0;
        #pragma unroll
        for (int r = 0; r < 8; ++r) {
            float s = fmaf(-2.0f, acc[r], cn[r]);
            if (s < best) { best = s; bidx = r + 8 * hi; }
        }
        // merge the two half-wave centroid groups (same point lives in lane L and L+16)
        float ob = __shfl_xor(best, 16, 32);
        int   oi = __shfl_xor(bidx, 16, 32);
        if (ob < best || (ob == best && oi < bidx)) { best = ob; bidx = oi; }

        // accumulate this point's 32 elements held by this lane into its cluster row
        float* row = lsum + bidx * SUMSTRIDE + 2 * hi;
        #pragma unroll
        for (int j = 0; j < 16; ++j) {
            atomicAdd(row + 4 * j,     b[j].x);
            atomicAdd(row + 4 * j + 1, b[j].y);
        }
        if (hi == 0) atomicAdd(&lcnt[bidx], 1.0f);
    }
    __syncthreads();

    // flush per-block partials with fp32 global atomics
    float* gsum = ws + WS_SUM;
    float* gcnt = ws + WS_CNT;
    for (int i = threadIdx.x; i < K_CL * EMBED; i += blockDim.x) {
        int mm = i >> 6, kk = i & 63;
        float v = lsum[mm * SUMSTRIDE + kk];
        if (v != 0.0f)
            __hip_atomic_fetch_add(&gsum[i], v, __ATOMIC_RELAXED, __HIP_MEMORY_SCOPE_AGENT);
    }
    for (int i = threadIdx.x; i < K_CL; i += blockDim.x) {
        float v = lcnt[i];
        if (v != 0.0f)
            __hip_atomic_fetch_add(&gcnt[i], v, __ATOMIC_RELAXED, __HIP_MEMORY_SCOPE_AGENT);
    }
}

__global__ void kmeans_update(float* __restrict__ ws, float* __restrict__ out) {
    int t = threadIdx.x;  // 1024 threads, 1 block
    float* cent  = ws + WS_CENT;
    float* cnorm = ws + WS_CNORM;
    float* gsum  = ws + WS_SUM;
    float* gcnt  = ws + WS_CNT;
    float c = 0.0f;
    if (t < K_CL * EMBED) {
        float cntv = gcnt[t >> 6];
        c = gsum[t] / fmaxf(cntv, 1.0f);
    }
    __syncthreads();                       // all accumulator reads done
    cent[t] = (t < K_CL * EMBED) ? c : 0.0f;
    if (t < K_CL * EMBED) { gsum[t] = 0.0f; out[t] = c; }
    if (t < K_CL)          gcnt[t] = 0.0f;
    __syncthreads();                       // padded centroids written
    if (t < 16) {
        float s = 0.0f;
        for (int k = 0; k < EMBED; ++k) { float v = cent[t * EMBED + k]; s = fmaf(v, v, s); }
        cnorm[t] = (t < K_CL) ? s : 3.0e38f;
    }
}

extern "C" void kernel_launch(void* const* d_in, const int* in_sizes, int n_in,
                              void* d_out, int out_size, void* d_ws, size_t ws_size,
                              hipStream_t stream) {
    const float* x  = (const float*)d_in[0];
    const float* c0 = (const float*)d_in[1];
    float* out = (float*)d_out;
    float* ws  = (float*)d_ws;

    const int npts   = in_sizes[0] / EMBED;   // 819200
    const int ntiles = npts / 16;             // 51200 (exact)
    const int blocks = 320;                   // 320*8 waves -> 20 tiles/wave exactly

    kmeans_init<<<1, 1024, 0, stream>>>(c0, ws);
    for (int it = 0; it < ITERS; ++it) {
        kmeans_assign<<<blocks, 256, 0, stream>>>(x, ws, ntiles);
        kmeans_update<<<1, 1024, 0, stream>>>(ws, out);
    }
}